// TimingPropagation_45329084842413
// MI455X (gfx1250) — compile-verified
//
#include <hip/hip_runtime.h>
#include <cstdint>
#include <cstddef>

// Timing-arc LUT bilinear interpolation for MI455X (gfx1250, wave32).
// Memory-bound: tables (~10.7 MB) stay L2-resident; streams use NT policy.

static __device__ __forceinline__ float lut_one(
    float qt, float qc, int idx, int dt, int dc,
    const float* __restrict__ values,
    const float* __restrict__ trans,
    const float* __restrict__ cap)
{
    constexpr float kEps = 1e-12f;

    const float* tr = trans + (size_t)idx * 8u;
    const float* cr = cap   + (size_t)idx * 8u;

    // Full padded rows for branchless searchsorted(right=True).
    const float4 ta = *reinterpret_cast<const float4*>(tr);
    const float4 tb = *reinterpret_cast<const float4*>(tr + 4);
    const float4 ca = *reinterpret_cast<const float4*>(cr);
    const float4 cb = *reinterpret_cast<const float4*>(cr + 4);

    int tri = (int)(ta.x <= qt) + (int)(ta.y <= qt) + (int)(ta.z <= qt) + (int)(ta.w <= qt)
            + (int)(tb.x <= qt) + (int)(tb.y <= qt) + (int)(tb.z <= qt) + (int)(tb.w <= qt);
    int cri = (int)(ca.x <= qc) + (int)(ca.y <= qc) + (int)(ca.z <= qc) + (int)(ca.w <= qc)
            + (int)(cb.x <= qc) + (int)(cb.y <= qc) + (int)(cb.z <= qc) + (int)(cb.w <= qc);

    const int max_t = max(dt - 1, 0);
    const int max_c = max(dc - 1, 0);
    const int ti  = min(max(tri, 1), max_t);
    const int ci  = min(max(cri, 1), max_c);
    const int tlo = max(ti - 1, 0);
    const int clo = max(ci - 1, 0);

    // Scalar re-reads of just-touched lines: guaranteed L0 hits, avoids
    // dynamic-indexed private arrays (scratch spills).
    const float t0 = tr[tlo];
    const float t1 = tr[ti];
    const float c0 = cr[clo];
    const float c1 = cr[ci];

    const float* vr = values + (size_t)idx * 64u;
    const float v00 = vr[tlo * 8 + clo];
    const float v01 = vr[tlo * 8 + ci];
    const float v10 = vr[ti  * 8 + clo];
    const float v11 = vr[ti  * 8 + ci];

    const float t_int = t1 - t0;
    const float c_int = c1 - c0;
    const bool  t_deg = fabsf(t_int) < kEps;
    const bool  c_deg = fabsf(c_int) < kEps;

    const float x = fminf(fmaxf(qt, t0), t1);
    const float y = fminf(fmaxf(qc, c0), c1);
    const float wa = (t1 - x) * (c1 - y);
    const float wb = (t1 - x) * (y - c0);
    const float wc = (x - t0) * (c1 - y);
    const float wd = (x - t0) * (y - c0);

    const float t_safe = t_deg ? kEps : t_int;
    const float c_safe = c_deg ? kEps : c_int;
    const float denom  = t_safe * c_safe;

    const float lc = fminf(fmaxf((y - c0) / fmaxf(c_safe, kEps), 0.0f), 1.0f);
    const float lt = fminf(fmaxf((x - t0) / fmaxf(t_safe, kEps), 0.0f), 1.0f);
    const float val_t_deg = v00 + lc * (v01 - v00);
    const float val_c_deg = v00 + lt * (v10 - v00);

    const bool  valid_den = fabsf(denom) >= kEps;
    const float denom_s   = valid_den ? denom : 1.0f;
    float bil = (v00 * wa + v01 * wb + v10 * wc + v11 * wd) / denom_s;
    bil = (valid_den && __builtin_isfinite(bil)) ? bil : v00;

    float out = (t_deg && c_deg) ? v00
              : (t_deg ? val_t_deg
              : (c_deg ? val_c_deg : bil));

    const bool valid_arc = (dt > 0) && (dc > 0);
    return (valid_arc && __builtin_isfinite(out)) ? out : 0.0f;
}

__global__ __launch_bounds__(256) void TimingPropagation_45329084842413_kernel(
    const float* __restrict__ in_rtrans,
    const float* __restrict__ in_ftrans,
    const float* __restrict__ out_caps,
    const int*   __restrict__ arc_idxs,
    const float* __restrict__ rd_v, const float* __restrict__ rd_t,
    const float* __restrict__ rd_c, const int*   __restrict__ rd_d,
    const float* __restrict__ fd_v, const float* __restrict__ fd_t,
    const float* __restrict__ fd_c, const int*   __restrict__ fd_d,
    const float* __restrict__ rt_v, const float* __restrict__ rt_t,
    const float* __restrict__ rt_c, const int*   __restrict__ rt_d,
    const float* __restrict__ ft_v, const float* __restrict__ ft_t,
    const float* __restrict__ ft_c, const int*   __restrict__ ft_d,
    float* __restrict__ out, int n)
{
    const int i = blockIdx.x * blockDim.x + threadIdx.x;
    if (i >= n) return;

    // Streaming inputs: non-temporal so they don't evict L2-resident LUTs.
    const float qr  = __builtin_nontemporal_load(in_rtrans + i);
    const float qf  = __builtin_nontemporal_load(in_ftrans + i);
    const float qc  = __builtin_nontemporal_load(out_caps  + i);
    const int   idx = __builtin_nontemporal_load(arc_idxs  + i);

    const size_t vrow = (size_t)idx * 64u;
    // Prefetch the four 256B values rows NOW (gfx1250 global_prefetch_b8):
    // the actual 4-point gathers depend on the searchsorted result, so this
    // hides the dependent-load latency behind the row loads + compares.
    __builtin_prefetch(rd_v + vrow, 0, 3);
    __builtin_prefetch(rd_v + vrow + 32, 0, 3);
    __builtin_prefetch(fd_v + vrow, 0, 3);
    __builtin_prefetch(fd_v + vrow + 32, 0, 3);
    __builtin_prefetch(rt_v + vrow, 0, 3);
    __builtin_prefetch(rt_v + vrow + 32, 0, 3);
    __builtin_prefetch(ft_v + vrow, 0, 3);
    __builtin_prefetch(ft_v + vrow + 32, 0, 3);

    const int2 d_rd = *reinterpret_cast<const int2*>(rd_d + (size_t)idx * 2u);
    const int2 d_fd = *reinterpret_cast<const int2*>(fd_d + (size_t)idx * 2u);
    const int2 d_rt = *reinterpret_cast<const int2*>(rt_d + (size_t)idx * 2u);
    const int2 d_ft = *reinterpret_cast<const int2*>(ft_d + (size_t)idx * 2u);

    const float rd = lut_one(qr, qc, idx, d_rd.x, d_rd.y, rd_v, rd_t, rd_c);
    const float fd = lut_one(qf, qc, idx, d_fd.x, d_fd.y, fd_v, fd_t, fd_c);
    const float rt = lut_one(qr, qc, idx, d_rt.x, d_rt.y, rt_v, rt_t, rt_c);
    const float ft = lut_one(qf, qc, idx, d_ft.x, d_ft.y, ft_v, ft_t, ft_c);

    // Output layout: [4, N] in stack order rd, fd, rt, ft. NT stores: outputs
    // are write-once, keep them out of the way of the LUT tables in L2.
    const size_t nn = (size_t)n;
    __builtin_nontemporal_store(rd, out + 0 * nn + i);
    __builtin_nontemporal_store(fd, out + 1 * nn + i);
    __builtin_nontemporal_store(rt, out + 2 * nn + i);
    __builtin_nontemporal_store(ft, out + 3 * nn + i);
}

extern "C" void kernel_launch(void* const* d_in, const int* in_sizes, int n_in,
                              void* d_out, int out_size, void* d_ws, size_t ws_size,
                              hipStream_t stream) {
    (void)n_in; (void)out_size; (void)d_ws; (void)ws_size;

    const float* in_rtrans = (const float*)d_in[0];
    const float* in_ftrans = (const float*)d_in[1];
    const float* out_caps  = (const float*)d_in[2];
    const int*   arc_idxs  = (const int*)  d_in[3];

    const float* rd_v = (const float*)d_in[4];
    const float* rd_t = (const float*)d_in[5];
    const float* rd_c = (const float*)d_in[6];
    const int*   rd_d = (const int*)  d_in[7];

    const float* fd_v = (const float*)d_in[8];
    const float* fd_t = (const float*)d_in[9];
    const float* fd_c = (const float*)d_in[10];
    const int*   fd_d = (const int*)  d_in[11];

    const float* rt_v = (const float*)d_in[12];
    const float* rt_t = (const float*)d_in[13];
    const float* rt_c = (const float*)d_in[14];
    const int*   rt_d = (const int*)  d_in[15];

    const float* ft_v = (const float*)d_in[16];
    const float* ft_t = (const float*)d_in[17];
    const float* ft_c = (const float*)d_in[18];
    const int*   ft_d = (const int*)  d_in[19];

    float* out = (float*)d_out;
    const int n = in_sizes[0];

    const int block = 256;            // 8 wave32s per block
    const int grid  = (n + block - 1) / block;

    TimingPropagation_45329084842413_kernel<<<grid, block, 0, stream>>>(
        in_rtrans, in_ftrans, out_caps, arc_idxs,
        rd_v, rd_t, rd_c, rd_d,
        fd_v, fd_t, fd_c, fd_d,
        rt_v, rt_t, rt_c, rt_d,
        ft_v, ft_t, ft_c, ft_d,
        out, n);
}